// DeltaModel_3204045603534
// MI455X (gfx1250) — compile-verified
//
#include <hip/hip_runtime.h>
#include <hip/hip_bf16.h>

// ---------------------------------------------------------------------------
// DeltaModel on MI455X (gfx1250, wave32).
//
// Phase A (token-parallel, trivial vs 23.3 TB/s): h = LN(emb + FFN(emb)) for
//   all B*L tokens -> f32 workspace. VALU, weights broadcast from LDS.
//
// Phase B (latency-bound scan): chunked WY-form delta rule, chunk C=16.
//   Per chunk: fused {G = K K^T, B0 = M K^T} via WMMA f32 16x16x4 (shared
//   K-fragment), 16-step triangular solve for U (VALU), M += U^T K (WMMA).
//   One workgroup per batch element; 4 waves, each owns a 16-row strip of M.
//   Chunk keys are DOUBLE-BUFFERED in LDS via GLOBAL_LOAD_ASYNC_TO_LDS_B128
//   (ASYNCcnt), so chunk c+1 streams in while chunk c is processed.
// ---------------------------------------------------------------------------

typedef __attribute__((ext_vector_type(2))) float v2f;
typedef __attribute__((ext_vector_type(8))) float v8f;
typedef int b128_t __attribute__((vector_size(16)));   // builtin's pointee type

#define H 64
#define TPB 128

#define AS1 __attribute__((address_space(1)))
#define AS3 __attribute__((address_space(3)))

#ifndef __has_builtin
#define __has_builtin(x) 0
#endif

#if __has_builtin(__builtin_amdgcn_global_load_async_to_lds_b128)
#define HAVE_ASYNC_LOAD 1
#else
#define HAVE_ASYNC_LOAD 0
#endif

#if __has_builtin(__builtin_amdgcn_s_wait_asynccnt)
#define WAIT_ASYNC(n) __builtin_amdgcn_s_wait_asynccnt(n)
#else
#define WAIT_ASYNC(n) asm volatile("s_wait_asynccnt %0" ::"n"(n) : "memory")
#endif

// ---------------------------------------------------------------------------
// Phase A: embedding + FFN + residual + LayerNorm
// ---------------------------------------------------------------------------
__global__ __launch_bounds__(TPB) void ffn_ln_kernel(
    const int* __restrict__ seq, const float* __restrict__ embed,
    const float* __restrict__ W1, const float* __restrict__ b1,
    const float* __restrict__ W2, const float* __restrict__ b2,
    const float* __restrict__ gamma, const float* __restrict__ beta,
    float* __restrict__ hws, int ntok)
{
    __shared__ float sE[H * H];        // 64x64 embedding table
    __shared__ float sW1[H * 2 * H];   // 64x128
    __shared__ float sW2[2 * H * H];   // 128x64
    __shared__ float sb1[2 * H], sb2[H], sg[H], sbt[H];

    const int tid = threadIdx.x;
    for (int i = tid; i < H * H; i += TPB)      sE[i]  = embed[i];
    for (int i = tid; i < H * 2 * H; i += TPB)  sW1[i] = W1[i];
    for (int i = tid; i < 2 * H * H; i += TPB)  sW2[i] = W2[i];
    sb1[tid] = b1[tid];
    if (tid < H) { sb2[tid] = b2[tid]; sg[tid] = gamma[tid]; sbt[tid] = beta[tid]; }
    __syncthreads();

    const int token = blockIdx.x * TPB + tid;
    if (token >= ntok) return;

    const int v = seq[token];
    float e[H], f[H];
    #pragma unroll
    for (int j = 0; j < H; ++j) { e[j] = sE[v * H + j]; f[j] = sb2[j]; }

    // ff = relu(e @ W1 + b1) @ W2 + b2 (weight reads wave-uniform -> LDS broadcast)
    for (int kk = 0; kk < 2 * H; ++kk) {
        float z = sb1[kk];
        #pragma unroll
        for (int j = 0; j < H; ++j) z = fmaf(e[j], sW1[j * 2 * H + kk], z);
        z = fmaxf(z, 0.0f);
        #pragma unroll
        for (int j = 0; j < H; ++j) f[j] = fmaf(z, sW2[kk * H + j], f[j]);
    }

    // x = e + f ; LayerNorm
    float s = 0.0f, s2 = 0.0f;
    #pragma unroll
    for (int j = 0; j < H; ++j) {
        float x = e[j] + f[j];
        f[j] = x;
        s += x; s2 = fmaf(x, x, s2);
    }
    const float mu  = s * (1.0f / H);
    const float var = s2 * (1.0f / H) - mu * mu;
    const float inv = rsqrtf(var + 1e-5f);
    #pragma unroll
    for (int j = 0; j < H; ++j)
        hws[(size_t)token * H + j] = (f[j] - mu) * inv * sg[j] + sbt[j];
}

// ---------------------------------------------------------------------------
// Async issue of one 16xH chunk of keys into an LDS buffer (2 x B128 / lane).
// Rows are 272B apart (H+4 floats) so every 16B transfer is aligned.
// ---------------------------------------------------------------------------
__device__ __forceinline__ void issue_chunk_async(
    const float* hb, int cc, float (*Kb)[H + 4], int tid)
{
#if HAVE_ASYNC_LOAD
    #pragma unroll
    for (int r = 0; r < 2; ++r) {
        const int e = (r * TPB + tid) * 4;      // 4 floats per lane per round
        const int t = e >> 6, j = e & 63;
        const float* src = hb + (size_t)(cc * 16 + t) * H + j;
        __builtin_amdgcn_global_load_async_to_lds_b128(
            (AS1 b128_t*)src, (AS3 b128_t*)&Kb[t][j], 0, 0);
    }
#else
    (void)hb; (void)cc; (void)Kb; (void)tid;
#endif
}

// ---------------------------------------------------------------------------
// Phase B: chunked delta-rule scan + output head. One block per batch element.
// ---------------------------------------------------------------------------
__global__ __launch_bounds__(TPB) void delta_scan_kernel(
    const float* __restrict__ hws,
    const float* __restrict__ Wr, const float* __restrict__ br,
    const float* __restrict__ Wo, const float* __restrict__ bo,
    float* __restrict__ out, int Lm1, int Ltot)
{
    __shared__ float Ms[H][H + 1];       // state M (column reads bank-spread)
    __shared__ float Ks[2][16][H + 4];   // double-buffered chunk keys, 16B rows
    __shared__ float Us[16][H + 1];      // chunk u vectors
    __shared__ float B0[H][17];          // B0[i][t] = (M0 k_t)_i
    __shared__ float Gs[16][17];         // Gram
    __shared__ float ctxs[H], tmps[H];

    const int tid  = threadIdx.x;
    const int b    = blockIdx.x;
    const int lane = tid & 31;
    const int w    = tid >> 5;        // wave id 0..3 -> owns M rows [16w,16w+16)
    const int m    = lane & 15;       // M-row / N-col within tile
    const int hi   = lane >> 4;       // half-wave select (K pair 0/1 vs 2/3)
    const int row  = 16 * w + m;

    const float* hb = hws + (size_t)b * Ltot * H;
    const int NC = (Lm1 + 15) >> 4;   // 128 chunks (last one has 15 steps)

#if HAVE_ASYNC_LOAD
    issue_chunk_async(hb, 0, Ks[0], tid);          // prologue: chunk 0 -> buf 0
#endif

    for (int i = tid; i < H * H; i += TPB) Ms[i >> 6][i & 63] = 0.0f;

    for (int c = 0; c < NC; ++c) {
        const int base = c * 16;
        const int clen = min(16, Lm1 - base);
#if HAVE_ASYNC_LOAD
        float (*Kb)[H + 4] = Ks[c & 1];
        if (c + 1 < NC) {
            issue_chunk_async(hb, c + 1, Ks[(c + 1) & 1], tid);
            WAIT_ASYNC(2);            // current chunk landed; next in flight
        } else {
            WAIT_ASYNC(0);
        }
        if (clen < 16) {              // zero-pad tail rows (only final chunk)
            for (int i = tid; i < (16 - clen) * H; i += TPB) {
                int t = clen + (i >> 6), j = i & 63;
                Kb[t][j] = 0.0f;
            }
        }
#else
        float (*Kb)[H + 4] = Ks[0];
        for (int i = tid; i < 16 * H; i += TPB) {
            int t = i >> 6, j = i & 63;
            Kb[t][j] = (t < clen) ? hb[(size_t)(base + t) * H + j] : 0.0f;
        }
        if (c + 1 < NC)
            __builtin_prefetch(hb + (size_t)(base + 16) * H + tid * 8, 0, 0);
#endif
        __syncthreads();

        // --- fused: G = K K^T  and  B0 = M K^T (wave's 16-row strip) -------
        // Shared B-fragment ka: B[K=j][N=t] = Ks[t][j]; also A-frag of Gram.
        v8f gacc = {};
        v8f acc  = {};
        for (int jj = 0; jj < H; jj += 4) {
            v2f ka, a;
            ka.x = Kb[m][jj + 2 * hi];
            ka.y = Kb[m][jj + 2 * hi + 1];
            a.x  = Ms[row][jj + 2 * hi];
            a.y  = Ms[row][jj + 2 * hi + 1];
            gacc = __builtin_amdgcn_wmma_f32_16x16x4_f32(
                false, ka, false, ka, (short)0, gacc, false, false);
            acc = __builtin_amdgcn_wmma_f32_16x16x4_f32(
                false, a, false, ka, (short)0, acc, false, false);
        }
        if (w == 0) {
            #pragma unroll
            for (int r = 0; r < 8; ++r) Gs[r + 8 * hi][m] = gacc[r];
        }
        #pragma unroll
        for (int r = 0; r < 8; ++r) B0[16 * w + r + 8 * hi][m] = acc[r];
        __syncthreads();

        // --- triangular solve: u_t = k_t - (B0_t + sum_{s<t} G[t][s] u_s)/d_t
        if (tid < H) {
            const int j = tid;
            for (int t = 0; t < 16; ++t) {
                float tv = B0[j][t];
                for (int sdx = 0; sdx < t; ++sdx) tv = fmaf(Gs[t][sdx], Us[sdx][j], tv);
                const float d = Gs[t][t] + 1e-6f;
                Us[t][j] = Kb[t][j] - tv / d;
            }
        }
        __syncthreads();

        // --- M += U^T K : 4 column tiles per wave, WMMA f32 16x16x4, K=16 ---
        for (int ct = 0; ct < 4; ++ct) {
            v8f cacc;
            #pragma unroll
            for (int r = 0; r < 8; ++r) cacc[r] = Ms[16 * w + r + 8 * hi][16 * ct + m];
            for (int tt = 0; tt < 16; tt += 4) {
                v2f a, bb;
                a.x  = Us[tt + 2 * hi][row];       // A[M=i][K=t] = U[t][i]
                a.y  = Us[tt + 2 * hi + 1][row];
                bb.x = Kb[tt + 2 * hi][16 * ct + m];
                bb.y = Kb[tt + 2 * hi + 1][16 * ct + m];
                cacc = __builtin_amdgcn_wmma_f32_16x16x4_f32(
                    false, a, false, bb, (short)0, cacc, false, false);
            }
            #pragma unroll
            for (int r = 0; r < 8; ++r) Ms[16 * w + r + 8 * hi][16 * ct + m] = cacc[r];
        }
        __syncthreads();
    }

    // --- ctx = M q ; out = (ctx @ Wr + br) @ Wo + bo ---
    if (tid < H) {
        const float* q = hb + (size_t)(Ltot - 1) * H;
        float s = 0.0f;
        for (int j = 0; j < H; ++j) s = fmaf(Ms[tid][j], q[j], s);
        ctxs[tid] = s;
    }
    __syncthreads();
    if (tid < H) {
        float s = br[tid];
        for (int j = 0; j < H; ++j) s = fmaf(ctxs[j], Wr[j * H + tid], s);
        tmps[tid] = s;
    }
    __syncthreads();
    if (tid < H) {
        float s = bo[tid];
        for (int j = 0; j < H; ++j) s = fmaf(tmps[j], Wo[j * H + tid], s);
        out[b * H + tid] = s;
    }
}

// ---------------------------------------------------------------------------
extern "C" void kernel_launch(void* const* d_in, const int* in_sizes, int n_in,
                              void* d_out, int out_size, void* d_ws, size_t ws_size,
                              hipStream_t stream) {
    const int*   seq   = (const int*)  d_in[0];
    const float* embed = (const float*)d_in[1];
    const float* W1    = (const float*)d_in[2];
    const float* b1    = (const float*)d_in[3];
    const float* W2    = (const float*)d_in[4];
    const float* b2    = (const float*)d_in[5];
    const float* gamma = (const float*)d_in[6];
    const float* beta  = (const float*)d_in[7];
    const float* Wr    = (const float*)d_in[8];
    const float* br    = (const float*)d_in[9];
    const float* Wo    = (const float*)d_in[10];
    const float* bo    = (const float*)d_in[11];

    const int ntok = in_sizes[0];          // B * L = 32768
    const int B    = 16;
    const int L    = ntok / B;             // 2048
    float* hws = (float*)d_ws;             // B*L*H f32 = 8 MB (L2-resident)

    ffn_ln_kernel<<<(ntok + TPB - 1) / TPB, TPB, 0, stream>>>(
        seq, embed, W1, b1, W2, b2, gamma, beta, hws, ntok);

    delta_scan_kernel<<<B, TPB, 0, stream>>>(
        hws, Wr, br, Wo, bo, (float*)d_out, L - 1, L);
}